// MFATBlock_53223234732403
// MI455X (gfx1250) — compile-verified
//
#include <hip/hip_runtime.h>
#include <math.h>

typedef float v2f __attribute__((ext_vector_type(2)));
typedef float v8f __attribute__((ext_vector_type(8)));

#define NPIX   65536          // 4*128*128
#define CCH    96
#define MAXK   192

// ---------------- input index map (setup_inputs dict insertion order) ------
enum {
  IN_X = 0,
  NORM1_W, NORM1_B, NORM2_W, NORM2_B,
  MLKA_NW, MLKA_NB, MLKA_SCALE, MLKA_PFW, MLKA_PFB, MLKA_PLW, MLKA_PLB,
  BR_BASE,                      // 8 entries per branch: w1,b1,w2,b2,w3,b3,xw,xb
  GSAU_NW = BR_BASE + 24, GSAU_NB, GSAU_SCALE, GSAU_C1W, GSAU_C1B,
  GSAU_DWW, GSAU_DWB, GSAU_C2W, GSAU_C2B,
  DFFN_PINW, DFFN_DWW, DFFN_FFT, DFFN_POUTW
};

// ---------------------------------------------------------------------------
__device__ __forceinline__ float wave_sum(float v) {
#pragma unroll
  for (int o = 16; o > 0; o >>= 1) v += __shfl_xor(v, o, 32);
  return v;
}

// LayerNorm over channels (96), then a second channels-first LN on the result.
// outY = ln_last(x) (kept as MLKA shortcut), outN = ln_cf(outY) (MLKA input).
__global__ void __launch_bounds__(256)
k_ln_double(const float* __restrict__ x,
            const float* __restrict__ w1, const float* __restrict__ b1,
            const float* __restrict__ w2, const float* __restrict__ b2,
            float* __restrict__ outY, float* __restrict__ outN) {
  int pix  = blockIdx.x * 8 + (threadIdx.x >> 5);
  int lane = threadIdx.x & 31;
  if (pix >= NPIX) return;
  const float* row = x + (size_t)pix * CCH;
  float v[3], y[3];
#pragma unroll
  for (int j = 0; j < 3; ++j) v[j] = row[lane + 32 * j];
  float u = wave_sum(v[0] + v[1] + v[2]) * (1.f / 96.f);
  float q = 0.f;
#pragma unroll
  for (int j = 0; j < 3; ++j) { float d = v[j] - u; q += d * d; }
  float r = rsqrtf(wave_sum(q) * (1.f / 96.f) + 1e-5f);
#pragma unroll
  for (int j = 0; j < 3; ++j) {
    int c = lane + 32 * j;
    y[j] = (v[j] - u) * r * w1[c] + b1[c];
  }
  float u2 = wave_sum(y[0] + y[1] + y[2]) * (1.f / 96.f);
  float q2 = 0.f;
#pragma unroll
  for (int j = 0; j < 3; ++j) { float d = y[j] - u2; q2 += d * d; }
  float r2 = rsqrtf(wave_sum(q2) * (1.f / 96.f) + 1e-6f);
#pragma unroll
  for (int j = 0; j < 3; ++j) {
    int c = lane + 32 * j;
    outY[(size_t)pix * CCH + c] = y[j];
    outN[(size_t)pix * CCH + c] = (y[j] - u2) * r2 * w2[c] + b2[c];
  }
}

// Single LayerNorm over channels with parametric eps.
__global__ void __launch_bounds__(256)
k_ln_single(const float* __restrict__ x,
            const float* __restrict__ w, const float* __restrict__ b,
            float eps, float* __restrict__ outN) {
  int pix  = blockIdx.x * 8 + (threadIdx.x >> 5);
  int lane = threadIdx.x & 31;
  if (pix >= NPIX) return;
  const float* row = x + (size_t)pix * CCH;
  float v[3];
#pragma unroll
  for (int j = 0; j < 3; ++j) v[j] = row[lane + 32 * j];
  float u = wave_sum(v[0] + v[1] + v[2]) * (1.f / 96.f);
  float q = 0.f;
#pragma unroll
  for (int j = 0; j < 3; ++j) { float d = v[j] - u; q += d * d; }
  float r = rsqrtf(wave_sum(q) * (1.f / 96.f) + eps);
#pragma unroll
  for (int j = 0; j < 3; ++j) {
    int c = lane + 32 * j;
    outN[(size_t)pix * CCH + c] = (v[j] - u) * r * w[c] + b[c];
  }
}

// ------------------- WMMA fp32 GEMM: O = A[M,K] * W[N,K]^T (+bias) ---------
// One wave -> 16x16 tile; block = 8 waves = 128 rows sharing one 16-col
// weight panel. B-panel staged global->LDS via GLOBAL_LOAD_ASYNC_TO_LDS_B128
// (ASYNCcnt), then read as ds_load_b64 fragments (row pad K+4 => 32 lanes hit
// 32 distinct banks for K in {32,96,192}). Grids divide M,N exactly: no early
// return, EXEC all-ones at every WMMA.
__global__ void __launch_bounds__(256)
k_gemm_wmma(const float* __restrict__ A, int lda, int acol,
            const float* __restrict__ W,            // [N, K] row-major
            const float* __restrict__ bias,
            float* __restrict__ O, int ldo, int ocol,
            int M, int N, int K) {
  __shared__ float sW[16 * (MAXK + 4)];
  const int tid   = threadIdx.x;
  const int nbase = blockIdx.y * 16;
  const int ldw   = K + 4;

  // ---- async stage: 16 rows x K floats of W into LDS (16B per lane/chunk)
  int nch = K >> 2;                                  // 16B chunks per row
  for (int t = tid; t < 16 * nch; t += 256) {
    int n = t / nch, c4 = (t - n * nch) << 2;
    unsigned     ldsa = (unsigned)(size_t)&sW[n * ldw + c4];
    const float* gp   = W + (size_t)(nbase + n) * K + c4;
    asm volatile("global_load_async_to_lds_b128 %0, %1, off"
                 :: "v"(ldsa), "v"(gp) : "memory");
  }
  asm volatile("s_wait_asynccnt 0" ::: "memory");
  __syncthreads();

  const int wave = tid >> 5, lane = tid & 31;
  const int mbase = (blockIdx.x * 8 + wave) * 16;
  const int l15 = lane & 15;
  const int ko  = (lane >> 4) * 2;                   // K sub-offset per half
  const float* arow = A + (size_t)(mbase + l15) * lda + acol;
  const float* brow = &sW[l15 * ldw];
  v8f c = {};
  for (int kk = 0; kk < K; kk += 4) {
    __builtin_prefetch(arow + kk + 64, 0, 1);
    v2f a = *(const v2f*)(arow + kk + ko);
    v2f b = *(const v2f*)(brow + kk + ko);
    c = __builtin_amdgcn_wmma_f32_16x16x4_f32(false, a, false, b,
                                              (short)0, c, false, false);
  }
  float bn = bias ? bias[nbase + l15] : 0.f;
  int   mo = mbase + ((lane >> 4) << 3);             // +0 or +8
  float* op = O + (size_t)mo * ldo + ocol + nbase + l15;
#pragma unroll
  for (int j = 0; j < 8; ++j) op[(size_t)j * ldo] = c[j] + bn;
}

// ------------------- depthwise conv (NHWC, H=W=128, zero pad) --------------
__global__ void __launch_bounds__(256)
k_dwconv(const float* __restrict__ in, int ldi,
         const float* __restrict__ wgt, const float* __restrict__ bias,
         float* __restrict__ out, int ldo,
         int cn, int ksz, int pad, int dil) {
  int idx = blockIdx.x * blockDim.x + threadIdx.x;
  if (idx >= NPIX * cn) return;
  int c = idx % cn, pix = idx / cn;
  int b = pix >> 14, rem = pix & 16383, yy = rem >> 7, xx = rem & 127;
  float acc = bias ? bias[c] : 0.f;
  const float* wc = wgt + c * ksz * ksz;
  for (int ky = 0; ky < ksz; ++ky) {
    int iy = yy + ky * dil - pad;
    if ((unsigned)iy >= 128u) continue;
    for (int kx = 0; kx < ksz; ++kx) {
      int ix = xx + kx * dil - pad;
      if ((unsigned)ix >= 128u) continue;
      acc += wc[ky * ksz + kx] *
             in[(size_t)((b << 14) + (iy << 7) + ix) * ldi + c];
    }
  }
  out[(size_t)pix * ldo + c] = acc;
}

// out[p,c] = pf[p, 96+c] * lka[p,c] * xb[p,c]   (MLKA gate, C wide)
__global__ void __launch_bounds__(256)
k_gate3(const float* __restrict__ pf, const float* __restrict__ lka,
        const float* __restrict__ xb, float* __restrict__ out) {
  int idx = blockIdx.x * blockDim.x + threadIdx.x;
  if (idx >= NPIX * CCH) return;
  int c = idx % CCH, p = idx / CCH;
  out[idx] = pf[(size_t)p * 192 + 96 + c] * lka[idx] * xb[idx];
}

// out[p,c] = a[p*lda+aofs+c] * b[p*ldb+bofs+c]
__global__ void __launch_bounds__(256)
k_mul2(const float* __restrict__ a, int lda, int aofs,
       const float* __restrict__ b, int ldb, int bofs,
       float* __restrict__ out, int ldo, int cn) {
  int idx = blockIdx.x * blockDim.x + threadIdx.x;
  if (idx >= NPIX * cn) return;
  int c = idx % cn, p = idx / cn;
  out[(size_t)p * ldo + c] =
      a[(size_t)p * lda + aofs + c] * b[(size_t)p * ldb + bofs + c];
}

// out = p*scale[c] + sc (+ extra)
__global__ void __launch_bounds__(256)
k_scale_res(const float* __restrict__ p, const float* __restrict__ scale,
            const float* __restrict__ sc, const float* __restrict__ extra,
            float* __restrict__ out) {
  int idx = blockIdx.x * blockDim.x + threadIdx.x;
  if (idx >= NPIX * CCH) return;
  int c = idx % CCH;
  float v = p[idx] * scale[c] + sc[idx];
  if (extra) v += extra[idx];
  out[idx] = v;
}

// ksp[c] = irfft2(F[c])  : 8x5 half-spectrum -> real 8x8 circular kernel
__global__ void k_build_ksp(const float* __restrict__ F, float* __restrict__ ksp) {
  int c = blockIdx.x, st = threadIdx.x;
  int s = st >> 3, t = st & 7;
  const float* Fc = F + c * 40;
  float acc = 0.f;
  for (int u = 0; u < 8; ++u)
    for (int v = 0; v < 8; ++v) {
      float f = (v <= 4) ? Fc[u * 5 + v] : Fc[((8 - u) & 7) * 5 + (8 - v)];
      acc += f * __cosf(0.78539816339f * (float)(u * s + v * t)); // 2*pi/8
    }
  ksp[c * 64 + st] = acc * (1.f / 64.f);
}

// per-channel 8x8 circular convolution inside each 8x8 patch (C2=384)
__global__ void __launch_bounds__(256)
k_circconv(const float* __restrict__ in, const float* __restrict__ ksp,
           float* __restrict__ out) {
  int idx = blockIdx.x * blockDim.x + threadIdx.x;
  if (idx >= NPIX * 384) return;
  int c = idx % 384, pix = idx / 384;
  int b = pix >> 14, rem = pix & 16383, yy = rem >> 7, xx = rem & 127;
  int py = yy & ~7, px = xx & ~7, u = yy & 7, v = xx & 7;
  const float* kc = ksp + c * 64;
  float acc = 0.f;
  for (int s = 0; s < 8; ++s) {
    int iy = py + ((u - s) & 7);
    for (int t = 0; t < 8; ++t) {
      int ix = px + ((v - t) & 7);
      acc += kc[s * 8 + t] *
             in[(size_t)((b << 14) + (iy << 7) + ix) * 384 + c];
    }
  }
  out[idx] = acc;
}

// h[p,c] = gelu_exact(x1[p,c]) * x2[p,c],  x1/x2 = big[:, 0:192 / 192:384]
__global__ void __launch_bounds__(256)
k_geglu(const float* __restrict__ big, float* __restrict__ h) {
  int idx = blockIdx.x * blockDim.x + threadIdx.x;
  if (idx >= NPIX * 192) return;
  int c = idx % 192, p = idx / 192;
  float x1 = big[(size_t)p * 384 + c];
  float x2 = big[(size_t)p * 384 + 192 + c];
  float g  = 0.5f * x1 * (1.f + erff(x1 * 0.70710678118f));
  h[idx] = g * x2;
}

__global__ void __launch_bounds__(256)
k_add_inplace(float* __restrict__ out, const float* __restrict__ a) {
  int idx = blockIdx.x * blockDim.x + threadIdx.x;
  if (idx >= NPIX * CCH) return;
  out[idx] += a[idx];
}

// ---------------------------------------------------------------------------
static inline dim3 ew_grid(long total) { return dim3((unsigned)((total + 255) / 256)); }

extern "C" void kernel_launch(void* const* d_in, const int* in_sizes, int n_in,
                              void* d_out, int out_size, void* d_ws, size_t ws_size,
                              hipStream_t stream) {
  const size_t MC = (size_t)NPIX * CCH;                 // 6,291,456 floats
  if (ws_size < (11 * MC + 384 * 64) * sizeof(float)) return;

  const float* xin = (const float*)d_in[IN_X];
  float* out = (float*)d_out;
  float* ws  = (float*)d_ws;

  float* t_y   = ws;               // 1 MC  : ln_last(x)        (MLKA shortcut)
  float* t_n   = ws + 1 * MC;      // 1 MC  : normed / gate temp
  float* t_pf  = ws + 2 * MC;      // 2 MC  : 2C tensors
  float* t_d1  = ws + 4 * MC;      // 1 MC
  float* t_d2  = ws + 5 * MC;      // 1 MC
  float* t_lka = ws + 6 * MC;      // 1 MC
  float* t_xb  = ws + 7 * MC;      // 1 MC
  float* t_x1  = ws + 8 * MC;      // 1 MC  : post-MLKA stream
  // DFFN phase aliases (sequential, no overlap with live data):
  float* t_n3   = ws + 8 * MC;     // reuse t_x1 slot after final residual
  float* t_big1 = ws;              // 4 MC (C2=384)
  float* t_big2 = ws + 4 * MC;     // 4 MC
  float* t_h    = ws + 8 * MC;     // 2 MC (HIDDEN=192)
  float* t_o    = ws + 10 * MC;    // 1 MC
  float* ksp    = ws + 11 * MC;    // 384*64

  dim3 lnG(NPIX / 8), blk(256);
  const int M = NPIX;

  // ---- norm1 + MLKA channels-first LN (fused double norm) ----
  k_ln_double<<<lnG, blk, 0, stream>>>(xin,
      (const float*)d_in[NORM1_W], (const float*)d_in[NORM1_B],
      (const float*)d_in[MLKA_NW], (const float*)d_in[MLKA_NB], t_y, t_n);

  // ---- MLKA: pf 1x1 (96 -> 192) ----
  k_gemm_wmma<<<dim3(M / 128, 192 / 16), blk, 0, stream>>>(
      t_n, CCH, 0, (const float*)d_in[MLKA_PFW], (const float*)d_in[MLKA_PFB],
      t_pf, 192, 0, M, 192, CCH);

  // ---- MLKA branches ----
  static const int K1[3] = {3, 5, 7}, P1[3] = {1, 2, 3};
  static const int K2[3] = {5, 7, 9}, D2[3] = {2, 3, 4}, P2[3] = {5, 9, 16};
  for (int i = 0; i < 3; ++i) {
    int c0 = 32 * i, bb = BR_BASE + 8 * i;
    const float* w1 = (const float*)d_in[bb + 0];
    const float* b1 = (const float*)d_in[bb + 1];
    const float* w2 = (const float*)d_in[bb + 2];
    const float* b2 = (const float*)d_in[bb + 3];
    const float* w3 = (const float*)d_in[bb + 4];
    const float* b3 = (const float*)d_in[bb + 5];
    const float* xw = (const float*)d_in[bb + 6];
    const float* xb = (const float*)d_in[bb + 7];
    dim3 g32 = ew_grid((long)NPIX * 32);
    k_dwconv<<<g32, blk, 0, stream>>>(t_pf + c0, 192, w1, b1, t_d1 + c0, CCH,
                                      32, K1[i], P1[i], 1);
    k_dwconv<<<g32, blk, 0, stream>>>(t_d1 + c0, CCH, w2, b2, t_d2 + c0, CCH,
                                      32, K2[i], P2[i], D2[i]);
    k_gemm_wmma<<<dim3(M / 128, 2), blk, 0, stream>>>(
        t_d2, CCH, c0, w3, b3, t_lka, CCH, c0, M, 32, 32);
    k_dwconv<<<g32, blk, 0, stream>>>(t_pf + c0, 192, xw, xb, t_xb + c0, CCH,
                                      32, K1[i], P1[i], 1);
  }

  // ---- gate: x * (lka * xb), then pl 1x1, scale + residual ----
  k_gate3<<<ew_grid((long)NPIX * CCH), blk, 0, stream>>>(t_pf, t_lka, t_xb, t_n);
  k_gemm_wmma<<<dim3(M / 128, 96 / 16), blk, 0, stream>>>(
      t_n, CCH, 0, (const float*)d_in[MLKA_PLW], (const float*)d_in[MLKA_PLB],
      t_d1, CCH, 0, M, CCH, CCH);
  k_scale_res<<<ew_grid((long)NPIX * CCH), blk, 0, stream>>>(
      t_d1, (const float*)d_in[MLKA_SCALE], t_y, (const float*)nullptr, t_x1);

  // ---- GSAU ----
  k_ln_single<<<lnG, blk, 0, stream>>>(t_x1,
      (const float*)d_in[GSAU_NW], (const float*)d_in[GSAU_NB], 1e-6f, t_n);
  k_gemm_wmma<<<dim3(M / 128, 192 / 16), blk, 0, stream>>>(
      t_n, CCH, 0, (const float*)d_in[GSAU_C1W], (const float*)d_in[GSAU_C1B],
      t_pf, 192, 0, M, 192, CCH);
  k_dwconv<<<ew_grid((long)NPIX * CCH), blk, 0, stream>>>(
      t_pf, 192, (const float*)d_in[GSAU_DWW], (const float*)d_in[GSAU_DWB],
      t_d2, CCH, CCH, 7, 3, 1);
  k_mul2<<<ew_grid((long)NPIX * CCH), blk, 0, stream>>>(
      t_pf, 192, 96, t_d2, CCH, 0, t_n, CCH, CCH);
  k_gemm_wmma<<<dim3(M / 128, 96 / 16), blk, 0, stream>>>(
      t_n, CCH, 0, (const float*)d_in[GSAU_C2W], (const float*)d_in[GSAU_C2B],
      t_lka, CCH, 0, M, CCH, CCH);
  // x = x_in + (c2*scale + t_x1)   -> running stream in d_out
  k_scale_res<<<ew_grid((long)NPIX * CCH), blk, 0, stream>>>(
      t_lka, (const float*)d_in[GSAU_SCALE], t_x1, xin, out);

  // ---- DFFN ----
  k_ln_single<<<lnG, blk, 0, stream>>>(out,
      (const float*)d_in[NORM2_W], (const float*)d_in[NORM2_B], 1e-5f, t_n3);
  k_gemm_wmma<<<dim3(M / 128, 384 / 16), blk, 0, stream>>>(
      t_n3, CCH, 0, (const float*)d_in[DFFN_PINW], (const float*)nullptr,
      t_big1, 384, 0, M, 384, CCH);
  k_build_ksp<<<dim3(384), dim3(64), 0, stream>>>((const float*)d_in[DFFN_FFT], ksp);
  k_circconv<<<ew_grid((long)NPIX * 384), blk, 0, stream>>>(t_big1, ksp, t_big2);
  k_dwconv<<<ew_grid((long)NPIX * 384), blk, 0, stream>>>(
      t_big2, 384, (const float*)d_in[DFFN_DWW], (const float*)nullptr,
      t_big1, 384, 384, 3, 1, 1);
  k_geglu<<<ew_grid((long)NPIX * 192), blk, 0, stream>>>(t_big1, t_h);
  k_gemm_wmma<<<dim3(M / 128, 96 / 16), blk, 0, stream>>>(
      t_h, 192, 0, (const float*)d_in[DFFN_POUTW], (const float*)nullptr,
      t_o, CCH, 0, M, CCH, 192);
  k_add_inplace<<<ew_grid((long)NPIX * CCH), blk, 0, stream>>>(out, t_o);
}